// CNF_42769284334097
// MI455X (gfx1250) — compile-verified
//
#include <hip/hip_runtime.h>
#include <math.h>

// ---------------------------------------------------------------------------
// CNF RK4 integrator for MI455X (gfx1250), fp32 WMMA path + async LDS staging.
//   vf:  inp(65) -> tanh(W1) -> 256 -> tanh(W2) -> 256 -> W3 -> 64
//   trace(J) = d1^T P d2  with  P[l,k] = W2[l,k] * (W3 @ W1[:64,:])[k,l]
// GEMMs use V_WMMA_F32_16X16X4_F32 (fp32 fidelity). W2/P k-slices are staged
// into LDS with GLOBAL_LOAD_ASYNC_TO_LDS_B128, double buffered on ASYNCcnt.
// ---------------------------------------------------------------------------

typedef float v2f __attribute__((ext_vector_type(2)));
typedef float v8f __attribute__((ext_vector_type(8)));

#define BATCH 4096
#define DDIM  64
#define HDIM  256
#define KP    68      // K for GEMM1: 64 (x) + 1 (t) padded to 68
#define BM    32      // batch rows per workgroup
#define LDH   260     // LDS stride for h buffers (260 % 64 == 4, conflict-free)
#define LDB   264     // LDS stride for staged B  (264 % 64 == 8, conflict-free)
#define NWAVE 8

// ---- gfx1250 async copy to LDS (ASYNCcnt) ---------------------------------
__device__ __forceinline__ void async_copy16(void* lds_dst, const void* gsrc) {
    uint32_t loff = (uint32_t)(uintptr_t)lds_dst;   // low 32 bits = LDS offset
    asm volatile("global_load_async_to_lds_b128 %0, %1, off"
                 :: "v"(loff), "v"(gsrc) : "memory");
}
__device__ __forceinline__ void wait_async16() {   // <=16 outstanding
    asm volatile("s_wait_asynccnt 0x10" ::: "memory");
}
__device__ __forceinline__ void wait_async0() {    // all done
    asm volatile("s_wait_asynccnt 0x0" ::: "memory");
}

// ---- WMMA fragment loaders (ISA 7.12.2 layouts, wave32) -------------------
__device__ __forceinline__ v2f ldA(const float* A, int lda, int mt, int k0, bool one_minus_sq) {
    const int lane = threadIdx.x & 31;
    const int m  = mt * 16 + (lane & 15);
    const int kk = k0 + ((lane >> 4) << 1);
    float a0 = A[m * lda + kk];
    float a1 = A[m * lda + kk + 1];
    v2f r;
    if (one_minus_sq) { r.x = 1.0f - a0 * a0; r.y = 1.0f - a1 * a1; }
    else              { r.x = a0;             r.y = a1; }
    return r;
}
__device__ __forceinline__ v2f ldB(const float* __restrict__ B, int ldb, int k0, int n0) {
    const int lane = threadIdx.x & 31;
    const int n  = n0 + (lane & 15);
    const int kk = k0 + ((lane >> 4) << 1);
    v2f r;
    r.x = B[kk * ldb + n];
    r.y = B[(kk + 1) * ldb + n];
    return r;
}
__device__ __forceinline__ v2f ldB_l(const float* B, int kk, int n0) {  // staged LDS B
    const int lane = threadIdx.x & 31;
    const int n = n0 + (lane & 15);
    const int k = kk + ((lane >> 4) << 1);
    v2f r;
    r.x = B[k * LDB + n];
    r.y = B[(k + 1) * LDB + n];
    return r;
}

// ---- generic GEMM with global B (used for GEMM1, K=68) --------------------
template <int KTOT, int NTOT, bool OTANH, bool BIAS>
__device__ __forceinline__ void gemm_to_lds(const float* Alds, int lda,
                                            const float* __restrict__ Bg,
                                            const float* __restrict__ bias,
                                            float* Dlds, int ldd) {
    const int lane = threadIdx.x & 31;
    const int wave = threadIdx.x >> 5;
    const int MT = BM / 16;
    const int NT = NTOT / 16;
    const int PAIRS = MT * (NT / 2);
    for (int p = wave; p < PAIRS; p += NWAVE) {
        const int mt  = p % MT;
        const int nt0 = (p / MT) * 2;
        float b0 = BIAS ? bias[nt0 * 16 + (lane & 15)]       : 0.0f;
        float b1 = BIAS ? bias[(nt0 + 1) * 16 + (lane & 15)] : 0.0f;
        v8f acc0 = {b0, b0, b0, b0, b0, b0, b0, b0};
        v8f acc1 = {b1, b1, b1, b1, b1, b1, b1, b1};
        for (int k = 0; k < KTOT; k += 4) {
            __builtin_prefetch(&Bg[(k + 16) * NTOT + nt0 * 16], 0, 1);
            v2f a   = ldA(Alds, lda, mt, k, false);
            v2f bb0 = ldB(Bg, NTOT, k, nt0 * 16);
            v2f bb1 = ldB(Bg, NTOT, k, nt0 * 16 + 16);
            acc0 = __builtin_amdgcn_wmma_f32_16x16x4_f32(false, a, false, bb0,
                                                         (short)0, acc0, false, false);
            acc1 = __builtin_amdgcn_wmma_f32_16x16x4_f32(false, a, false, bb1,
                                                         (short)0, acc1, false, false);
        }
        const int nb = nt0 * 16 + (lane & 15);
        const int mb = mt * 16 + ((lane >> 4) << 3);
        for (int r = 0; r < 8; ++r) {
            float v0 = acc0[r], v1 = acc1[r];
            if (OTANH) { v0 = tanhf(v0); v1 = tanhf(v1); }
            Dlds[(mb + r) * ldd + nb]      = v0;
            Dlds[(mb + r) * ldd + nb + 16] = v1;
        }
    }
}

// ---- fused vector-field evaluation + RK4 stage epilogue -------------------
__global__ __launch_bounds__(256) void cnf_vf_eval(
    const float* __restrict__ xsrc,
    float* y_x, float* y_ld, float* accx, float* accld, float* xs,
    const float* __restrict__ W1p, const float* __restrict__ b1,
    const float* __restrict__ W2,  const float* __restrict__ b2,
    const float* __restrict__ W3,  const float* __restrict__ b3,
    const float* __restrict__ Pm,  const float* __restrict__ tspan,
    int step, int stage, int last, float* outz, float* outld)
{
    __shared__ float xin[BM * KP];
    __shared__ float h1 [BM * LDH];
    __shared__ float h2 [BM * LDH];
    __shared__ float sW [2][BM * LDB];   // staged W2 slices; sW[0] reused as q
    __shared__ float sP [2][BM * LDB];   // staged P slices
    __shared__ float tred[BM * 8];
    __shared__ float trl [BM];

    const int tid  = threadIdx.x;
    const int lane = tid & 31;
    const int wave = tid >> 5;
    const int row0 = blockIdx.x * BM;

    const float t0 = tspan[step];
    const float dt = tspan[step + 1] - t0;
    const float cs = (stage == 0) ? 0.0f : (stage == 3 ? 1.0f : 0.5f);
    const float tc = t0 + cs * dt;

    // --- Phase A: async-stage x tile + first W2/P slice (overlaps GEMM1) ---
    for (int u = tid; u < BM * 16; u += 256) {          // x cols 0..63, b128
        const int m = u >> 4, c4 = (u & 15) << 2;
        async_copy16(&xin[m * KP + c4], &xsrc[(row0 + m) * DDIM + c4]);
    }
    for (int u = tid; u < 32 * 64; u += 256) {          // W2/P slice 0
        const int rr = u >> 6, c4 = (u & 63) << 2;
        async_copy16(&sW[0][rr * LDB + c4], &W2[rr * HDIM + c4]);
        async_copy16(&sP[0][rr * LDB + c4], &Pm[rr * HDIM + c4]);
    }
    wait_async16();                                     // x tile (issued first) done
    if (tid < BM) {                                     // [t | 0 0 0] columns
        xin[tid * KP + DDIM]     = tc;
        xin[tid * KP + DDIM + 1] = 0.0f;
        xin[tid * KP + DDIM + 2] = 0.0f;
        xin[tid * KP + DDIM + 3] = 0.0f;
    }
    __syncthreads();

    // --- GEMM1: h1 = tanh(inp @ W1p + b1) (slice-0 async loads in flight) --
    gemm_to_lds<KP, HDIM, true, true>(xin, KP, W1p, b1, h1, LDH);
    __syncthreads();

    // --- GEMM2 + trace GEMM, fused, async double-buffered B staging --------
    {
        const int mt  = wave & 1;
        const int nt0 = (wave >> 1) << 1;                 // 0,2,4,6
        const int nc0 = nt0 * 16, nc1 = nc0 + 16;
        const int nc2 = (nt0 + 8) * 16, nc3 = nc2 + 16;
        v8f a2[4], aq[4];
        {
            float bb0 = b2[nc0 + (lane & 15)], bb1 = b2[nc1 + (lane & 15)];
            float bb2 = b2[nc2 + (lane & 15)], bb3 = b2[nc3 + (lane & 15)];
            a2[0] = (v8f){bb0,bb0,bb0,bb0,bb0,bb0,bb0,bb0};
            a2[1] = (v8f){bb1,bb1,bb1,bb1,bb1,bb1,bb1,bb1};
            a2[2] = (v8f){bb2,bb2,bb2,bb2,bb2,bb2,bb2,bb2};
            a2[3] = (v8f){bb3,bb3,bb3,bb3,bb3,bb3,bb3,bb3};
            aq[0] = (v8f){0,0,0,0,0,0,0,0}; aq[1] = aq[0];
            aq[2] = aq[0];                  aq[3] = aq[0];
        }
        for (int s = 0; s < 8; ++s) {
            if (s < 7) {                                  // issue slice s+1
                float*       dW = sW[(s + 1) & 1];
                float*       dP = sP[(s + 1) & 1];
                const int    gs = (s + 1) * 32;
                for (int u = tid; u < 32 * 64; u += 256) {
                    const int rr = u >> 6, c4 = (u & 63) << 2;
                    async_copy16(&dW[rr * LDB + c4], &W2[(gs + rr) * HDIM + c4]);
                    async_copy16(&dP[rr * LDB + c4], &Pm[(gs + rr) * HDIM + c4]);
                }
                wait_async16();                           // slice s complete
            } else {
                wait_async0();
            }
            __syncthreads();
            const float* Wb = sW[s & 1];
            const float* Pb = sP[s & 1];
            for (int kk = 0; kk < 32; kk += 4) {
                v2f h = ldA(h1, LDH, mt, s * 32 + kk, false);
                v2f d; d.x = 1.0f - h.x * h.x; d.y = 1.0f - h.y * h.y;
                v2f w, p;
                w = ldB_l(Wb, kk, nc0);
                a2[0] = __builtin_amdgcn_wmma_f32_16x16x4_f32(false, h, false, w, (short)0, a2[0], false, false);
                p = ldB_l(Pb, kk, nc0);
                aq[0] = __builtin_amdgcn_wmma_f32_16x16x4_f32(false, d, false, p, (short)0, aq[0], false, false);
                w = ldB_l(Wb, kk, nc1);
                a2[1] = __builtin_amdgcn_wmma_f32_16x16x4_f32(false, h, false, w, (short)0, a2[1], false, false);
                p = ldB_l(Pb, kk, nc1);
                aq[1] = __builtin_amdgcn_wmma_f32_16x16x4_f32(false, d, false, p, (short)0, aq[1], false, false);
                w = ldB_l(Wb, kk, nc2);
                a2[2] = __builtin_amdgcn_wmma_f32_16x16x4_f32(false, h, false, w, (short)0, a2[2], false, false);
                p = ldB_l(Pb, kk, nc2);
                aq[2] = __builtin_amdgcn_wmma_f32_16x16x4_f32(false, d, false, p, (short)0, aq[2], false, false);
                w = ldB_l(Wb, kk, nc3);
                a2[3] = __builtin_amdgcn_wmma_f32_16x16x4_f32(false, h, false, w, (short)0, a2[3], false, false);
                p = ldB_l(Pb, kk, nc3);
                aq[3] = __builtin_amdgcn_wmma_f32_16x16x4_f32(false, d, false, p, (short)0, aq[3], false, false);
            }
            __syncthreads();
        }
        // store: h2 = tanh(a2); q (into sW[0], stride LDB) = aq
        float* qb = &sW[0][0];
        const int mb = mt * 16 + ((lane >> 4) << 3);
        const int ncs[4] = {nc0, nc1, nc2, nc3};
        for (int q = 0; q < 4; ++q) {
            const int nb = ncs[q] + (lane & 15);
            for (int r = 0; r < 8; ++r) {
                h2[(mb + r) * LDH + nb] = tanhf(a2[q][r]);
                qb[(mb + r) * LDB + nb] = aq[q][r];
            }
        }
    }
    __syncthreads();

    // --- trace[m] = sum_k q[m,k] * (1 - h2[m,k]^2) -------------------------
    {
        const float* qb = &sW[0][0];
        const int m = tid & (BM - 1);
        const int chunk = tid >> 5;
        float sacc = 0.0f;
        for (int k = chunk * 32; k < chunk * 32 + 32; ++k) {
            const float h = h2[m * LDH + k];
            sacc += qb[m * LDB + k] * (1.0f - h * h);
        }
        tred[m * 8 + chunk] = sacc;
    }
    __syncthreads();
    if (tid < BM) {
        float sacc = 0.0f;
        for (int c = 0; c < 8; ++c) sacc += tred[tid * 8 + c];
        trl[tid] = sacc;
    }
    __syncthreads();

    // --- GEMM3: out = h2 @ W3 + b3, fused RK4 stage epilogue ---------------
    {
        const int MT = 2, NT = DDIM / 16;
        for (int p = wave; p < MT * NT; p += NWAVE) {
            const int mt = p % MT, nt = p / MT;
            const float b = b3[nt * 16 + (lane & 15)];
            v8f acc = {b, b, b, b, b, b, b, b};
            for (int k = 0; k < HDIM; k += 4) {
                __builtin_prefetch(&W3[(k + 16) * DDIM + nt * 16], 0, 1);
                v2f a  = ldA(h2, LDH, mt, k, false);
                v2f bb = ldB(W3, DDIM, k, nt * 16);
                acc = __builtin_amdgcn_wmma_f32_16x16x4_f32(false, a, false, bb,
                                                            (short)0, acc, false, false);
            }
            const int n  = nt * 16 + (lane & 15);
            const int ml = mt * 16 + ((lane >> 4) << 3);
            for (int r = 0; r < 8; ++r) {
                const int gi = (row0 + ml + r) * DDIM + n;
                const float kx = acc[r];
                if (stage == 0) {
                    accx[gi] = kx;
                    xs[gi] = y_x[gi] + 0.5f * dt * kx;
                } else if (stage == 1) {
                    accx[gi] += 2.0f * kx;
                    xs[gi] = y_x[gi] + 0.5f * dt * kx;
                } else if (stage == 2) {
                    accx[gi] += 2.0f * kx;
                    xs[gi] = y_x[gi] + dt * kx;
                } else {
                    const float yn = y_x[gi] + (dt / 6.0f) * (accx[gi] + kx);
                    y_x[gi] = yn;
                    if (last) outz[gi] = yn;
                }
            }
        }
    }

    // --- log-det part ------------------------------------------------------
    if (tid < BM) {
        const int row = row0 + tid;
        const float kld = -trl[tid];
        if (stage == 0)      accld[row] = kld;
        else if (stage < 3)  accld[row] += 2.0f * kld;
        else {
            const float yn = y_ld[row] + (dt / 6.0f) * (accld[row] + kld);
            y_ld[row] = yn;
            if (last) outld[row] = yn;
        }
    }
}

// ---- prep kernels ---------------------------------------------------------
__global__ void cnf_prep_w1p(const float* __restrict__ W1, float* W1p) {
    const int i = blockIdx.x * blockDim.x + threadIdx.x;
    if (i < KP * HDIM) {
        const int r = i / HDIM;
        W1p[i] = (r < DDIM + 1) ? W1[i] : 0.0f;
    }
}
__global__ void cnf_prep_P(const float* __restrict__ W1, const float* __restrict__ W2,
                           const float* __restrict__ W3, float* Pm) {
    const int i = blockIdx.x * blockDim.x + threadIdx.x; // i = l*H + k
    if (i < HDIM * HDIM) {
        const int l = i / HDIM, k = i % HDIM;
        float s = 0.0f;
        for (int j = 0; j < DDIM; ++j) s += W3[k * DDIM + j] * W1[j * HDIM + l];
        Pm[i] = W2[i] * s;
    }
}
__global__ void cnf_init(const float* __restrict__ x, float* y_x, float* y_ld) {
    const int i = blockIdx.x * blockDim.x + threadIdx.x;
    if (i < BATCH * DDIM) y_x[i] = x[i];
    if (i < BATCH)        y_ld[i] = 0.0f;
}

// ---- host side ------------------------------------------------------------
extern "C" void kernel_launch(void* const* d_in, const int* in_sizes, int n_in,
                              void* d_out, int out_size, void* d_ws, size_t ws_size,
                              hipStream_t stream) {
    const float* x     = (const float*)d_in[0];
    const float* tspan = (const float*)d_in[1];
    const float* W1    = (const float*)d_in[2];
    const float* b1    = (const float*)d_in[3];
    const float* W2    = (const float*)d_in[4];
    const float* b2    = (const float*)d_in[5];
    const float* W3    = (const float*)d_in[6];
    const float* b3    = (const float*)d_in[7];

    float* ws = (float*)d_ws;
    float* y_x   = ws;
    float* y_ld  = y_x   + BATCH * DDIM;
    float* xs    = y_ld  + BATCH;
    float* accx  = xs    + BATCH * DDIM;
    float* accld = accx  + BATCH * DDIM;
    float* W1p   = accld + BATCH;
    float* Pm    = W1p   + KP * HDIM;

    float* outz  = (float*)d_out;
    float* outld = outz + BATCH * DDIM;

    cnf_prep_w1p<<<(KP * HDIM + 255) / 256, 256, 0, stream>>>(W1, W1p);
    cnf_prep_P<<<(HDIM * HDIM + 255) / 256, 256, 0, stream>>>(W1, W2, W3, Pm);
    cnf_init<<<(BATCH * DDIM + 255) / 256, 256, 0, stream>>>(x, y_x, y_ld);

    for (int step = 0; step < 16; ++step) {
        for (int stage = 0; stage < 4; ++stage) {
            const float* src = (stage == 0) ? y_x : xs;
            const int last = (step == 15 && stage == 3) ? 1 : 0;
            cnf_vf_eval<<<BATCH / BM, 256, 0, stream>>>(
                src, y_x, y_ld, accx, accld, xs,
                W1p, b1, W2, b2, W3, b3, Pm, tspan,
                step, stage, last, outz, outld);
        }
    }
}